// TGNLayerGraphSumEmbedding_21174188769660
// MI455X (gfx1250) — compile-verified
//
#include <hip/hip_runtime.h>

typedef float v2f __attribute__((ext_vector_type(2)));
typedef float v8f __attribute__((ext_vector_type(8)));

#define N_NODES 100000
#define K_NBR   16
#define DIM     128
#define DE_DIM  32
#define DT_DIM  32
#define D1      (DIM + DE_DIM + DT_DIM)   // 192
#define D2      (2 * DIM)                 // 256
#define M_TILE  32                        // 2 strips of 16 rows
#define AGG_PITCH 196   // 196 % 64 == 4 -> conflict-free row-indexed LDS reads
#define X2_PITCH  260   // 260 % 64 == 4

// Repack row-major W[K][128] into K-paired Wp[K/2][128] of float2 so each
// lane's WMMA B operand {W[k][c], W[k+1][c]} is one aligned b64 load.
__global__ __launch_bounds__(256)
void repack_w(const float* __restrict__ W, float2* __restrict__ Wp, int n_pairs_x_cols) {
    const int i = blockIdx.x * blockDim.x + threadIdx.x;
    if (i >= n_pairs_x_cols) return;
    const int col = i & (DIM - 1);
    const int kp  = i >> 7;
    Wp[i] = make_float2(W[(2 * kp) * DIM + col], W[(2 * kp + 1) * DIM + col]);
}

// One TGN layer: out[n] = concat(emb[n], relu(agg[n] @ W1 + b1)) @ W2 + b2
// agg[n] = [ sum_j emb[idx[n][j]] , sum_j edge[n][j] , sum_j time[n][j] ]
__global__ __launch_bounds__(128)
void tgn_layer(const float* __restrict__ emb,      // [N,128] current embeddings
               float* __restrict__ out,            // [N,128]
               const float* __restrict__ edge,     // [N,K,32] (this layer)
               const float* __restrict__ timef,    // [N,K,32] (this layer)
               const v2f* __restrict__ Wp1,        // [96,128]  K-paired
               const float* __restrict__ b1,       // [128]
               const v2f* __restrict__ Wp2,        // [128,128] K-paired
               const float* __restrict__ b2,       // [128]
               const long long* __restrict__ idx)  // [N,K] int64
{
    __shared__ float s_agg[M_TILE][AGG_PITCH];  // [nbr_sum(128) | e_sum(32) | t_sum(32)]
    __shared__ float s_x2[M_TILE][X2_PITCH];    // [emb row (128) | h (128)]

    const int tid   = threadIdx.x;
    const int lane  = tid & 31;
    const int wave  = tid >> 5;
    const int node0 = blockIdx.x * M_TILE;

    // ---------------- Phase 1: build agg + stage emb rows (wave handles 8 nodes)
    for (int mi = 0; mi < 8; ++mi) {
        const int m    = wave * 8 + mi;
        const int node = node0 + m;

        // neighbor gather-sum: lane covers dims [4*lane, 4*lane+4)
        const long long* nidx = idx + (long long)node * K_NBR;
        float4 acc = make_float4(0.f, 0.f, 0.f, 0.f);
#pragma unroll
        for (int j = 0; j < K_NBR; ++j) {
            const long long nb = nidx[j];  // wave-uniform -> saddr form
            const float4 v = *reinterpret_cast<const float4*>(emb + nb * DIM + lane * 4);
            acc.x += v.x; acc.y += v.y; acc.z += v.z; acc.w += v.w;
        }
        *reinterpret_cast<float4*>(&s_agg[m][lane * 4]) = acc;

        // edge/time sums: lane == feature dim (32); pure stream -> non-temporal
        const float* ep = edge  + ((long long)node * K_NBR) * DE_DIM + lane;
        const float* tp = timef + ((long long)node * K_NBR) * DT_DIM + lane;
        __builtin_prefetch(ep + K_NBR * DE_DIM, 0, 0);   // global_prefetch_b8 (speculative)
        __builtin_prefetch(tp + K_NBR * DT_DIM, 0, 0);
        float esum = 0.f, tsum = 0.f;
#pragma unroll
        for (int j = 0; j < K_NBR; ++j) {
            esum += __builtin_nontemporal_load(ep + j * DE_DIM);
            tsum += __builtin_nontemporal_load(tp + j * DT_DIM);
        }
        s_agg[m][DIM + lane]          = esum;
        s_agg[m][DIM + DE_DIM + lane] = tsum;

        // stage this node's own embedding row for the concat input of GEMM2
        const float4 me = *reinterpret_cast<const float4*>(emb + (long long)node * DIM + lane * 4);
        *reinterpret_cast<float4*>(&s_x2[m][lane * 4]) = me;
    }
    __syncthreads();

    // WMMA lane mapping (f32 16x16x4):
    //   A/B: lanes 0-15 hold K=k0,k0+1 ; lanes 16-31 hold K=k0+2,k0+3
    //   C/D: VGPR r holds row (lane>=16 ? 8 : 0)+r, col = lane&15
    const int mrow  = lane & 15;
    const int khalf = (lane >> 4);          // 0 or 1 (K-pair select)
    const int c0    = wave * 32 + mrow;     // this wave's two 16-col tiles
    const int c1    = c0 + 16;

    // ---------------- Phase 2: h = relu(agg @ W1 + b1) -> s_x2[:,128:256]
    v8f a00, a01, a10, a11;
#pragma unroll
    for (int i = 0; i < 8; ++i) { a00[i] = 0.f; a01[i] = 0.f; a10[i] = 0.f; a11[i] = 0.f; }

    for (int k0 = 0; k0 < D1; k0 += 4) {
        const int kp = (k0 >> 1) + khalf;                 // K-pair index
        v2f av0 = *reinterpret_cast<const v2f*>(&s_agg[mrow][k0 + khalf * 2]);
        v2f av1 = *reinterpret_cast<const v2f*>(&s_agg[16 + mrow][k0 + khalf * 2]);
        v2f bv0 = Wp1[kp * DIM + c0];                     // one b64 load per B tile
        v2f bv1 = Wp1[kp * DIM + c1];
        a00 = __builtin_amdgcn_wmma_f32_16x16x4_f32(false, av0, false, bv0, (short)0, a00, false, false);
        a01 = __builtin_amdgcn_wmma_f32_16x16x4_f32(false, av0, false, bv1, (short)0, a01, false, false);
        a10 = __builtin_amdgcn_wmma_f32_16x16x4_f32(false, av1, false, bv0, (short)0, a10, false, false);
        a11 = __builtin_amdgcn_wmma_f32_16x16x4_f32(false, av1, false, bv1, (short)0, a11, false, false);
    }

    // bias + relu, scatter accumulators into LDS h region
    {
        const float bb0 = b1[c0], bb1 = b1[c1];
        const int mbase = khalf * 8;
#pragma unroll
        for (int r = 0; r < 8; ++r) {
            s_x2[mbase + r][DIM + c0]      = fmaxf(a00[r] + bb0, 0.f);
            s_x2[mbase + r][DIM + c1]      = fmaxf(a01[r] + bb1, 0.f);
            s_x2[16 + mbase + r][DIM + c0] = fmaxf(a10[r] + bb0, 0.f);
            s_x2[16 + mbase + r][DIM + c1] = fmaxf(a11[r] + bb1, 0.f);
        }
    }
    __syncthreads();

    // ---------------- Phase 3: out = [emb, h] @ W2 + b2
#pragma unroll
    for (int i = 0; i < 8; ++i) { a00[i] = 0.f; a01[i] = 0.f; a10[i] = 0.f; a11[i] = 0.f; }

    for (int k0 = 0; k0 < D2; k0 += 4) {
        const int kp = (k0 >> 1) + khalf;
        v2f av0 = *reinterpret_cast<const v2f*>(&s_x2[mrow][k0 + khalf * 2]);
        v2f av1 = *reinterpret_cast<const v2f*>(&s_x2[16 + mrow][k0 + khalf * 2]);
        v2f bv0 = Wp2[kp * DIM + c0];
        v2f bv1 = Wp2[kp * DIM + c1];
        a00 = __builtin_amdgcn_wmma_f32_16x16x4_f32(false, av0, false, bv0, (short)0, a00, false, false);
        a01 = __builtin_amdgcn_wmma_f32_16x16x4_f32(false, av0, false, bv1, (short)0, a01, false, false);
        a10 = __builtin_amdgcn_wmma_f32_16x16x4_f32(false, av1, false, bv0, (short)0, a10, false, false);
        a11 = __builtin_amdgcn_wmma_f32_16x16x4_f32(false, av1, false, bv1, (short)0, a11, false, false);
    }

    {
        const float bb0 = b2[c0], bb1 = b2[c1];
        const int mbase = khalf * 8;
#pragma unroll
        for (int r = 0; r < 8; ++r) {
            const long long r0 = node0 + mbase + r;
            const long long r1 = r0 + 16;
            out[r0 * DIM + c0] = a00[r] + bb0;
            out[r0 * DIM + c1] = a01[r] + bb1;
            out[r1 * DIM + c0] = a10[r] + bb0;
            out[r1 * DIM + c1] = a11[r] + bb1;
        }
    }
}

extern "C" void kernel_launch(void* const* d_in, const int* in_sizes, int n_in,
                              void* d_out, int out_size, void* d_ws, size_t ws_size,
                              hipStream_t stream) {
    const float*     features   = (const float*)d_in[0];
    const float*     edge_feats = (const float*)d_in[1];   // [L,N,K,32]
    const float*     time_feats = (const float*)d_in[2];   // [L,N,K,32]
    const float*     W1         = (const float*)d_in[3];   // [L,192,128]
    const float*     b1         = (const float*)d_in[4];   // [L,128]
    const float*     W2         = (const float*)d_in[5];   // [L,256,128]
    const float*     b2         = (const float*)d_in[6];   // [L,128]
    const long long* nidx       = (const long long*)d_in[7]; // [L,N,K] int64 per reference
    float* out = (float*)d_out;

    // Workspace layout: [ tmp: N*128 f32 | Wp1: L*96*128 float2 | Wp2: L*128*128 float2 ]
    char*   ws  = (char*)d_ws;
    float*  tmp = (float*)ws;                                  // 51.2 MB
    float2* wp1 = (float2*)(ws + (size_t)N_NODES * DIM * 4);   // 192 KB
    float2* wp2 = wp1 + (size_t)2 * (D1 / 2) * DIM;            // 256 KB

    // Repack both layers' weights into K-paired float2 layout.
    const int n1 = 2 * (D1 / 2) * DIM;   // 24576
    const int n2 = 2 * (D2 / 2) * DIM;   // 32768
    repack_w<<<(n1 + 255) / 256, 256, 0, stream>>>(W1, wp1, n1);
    repack_w<<<(n2 + 255) / 256, 256, 0, stream>>>(W2, wp2, n2);

    const size_t EN = (size_t)N_NODES * K_NBR;
    dim3 grid(N_NODES / M_TILE), block(128);

    // layer 0: features -> tmp
    tgn_layer<<<grid, block, 0, stream>>>(features, tmp,
        edge_feats, time_feats,
        (const v2f*)wp1, b1, (const v2f*)wp2, b2, nidx);

    // layer 1: tmp -> out
    tgn_layer<<<grid, block, 0, stream>>>(tmp, out,
        edge_feats + EN * DE_DIM, time_feats + EN * DT_DIM,
        (const v2f*)(wp1 + (size_t)(D1 / 2) * DIM), b1 + DIM,
        (const v2f*)(wp2 + (size_t)(D2 / 2) * DIM), b2 + DIM,
        nidx + EN);
}